// ClassificationLoss_16587163697678
// MI455X (gfx1250) — compile-verified
//
#include <hip/hip_runtime.h>
#include <math.h>

typedef __attribute__((ext_vector_type(2))) float v2f;
typedef __attribute__((ext_vector_type(8))) float v8f;

#define B_TOT   512
#define M_TOT   4096
#define T_STEPS 48
#define M_TILES (M_TOT / 16)                // 256 tiles of 16 columns
#define CHUNKS  16                          // gridDim.x: M split into 16 chunks
#define TILES_PER_CHUNK (M_TILES / CHUNKS)  // 16 tiles per chunk
#define WAVES_PER_BLOCK 8

#define BOP_OFFSET 8192
#define BOP_ELEMS  ((size_t)M_TILES * T_STEPS * 32)   // float2 entries

// ---------------------------------------------------------------------------
// Kernel 0: init packed argmin slots to +inf
// ---------------------------------------------------------------------------
__global__ void cls_init_kernel(unsigned long long* __restrict__ packed) {
    int i = blockIdx.x * blockDim.x + threadIdx.x;
    if (i < B_TOT) packed[i] = 0xFFFFFFFFFFFFFFFFULL;
}

// ---------------------------------------------------------------------------
// Kernel 0b: precompute per-lane B operands for the WMMA.
//   Bop[(tile*48 + t)*32 + lane] = lane<16 ? {m0, m1} : {1, ||m||^2}
//   (column m = tile*16 + (lane&15)); coalesced 256B per (tile,t) per wave.
// ---------------------------------------------------------------------------
__global__ __launch_bounds__(256)
void cls_bop_kernel(const float* __restrict__ memory,
                    float2* __restrict__ bop) {
    int idx = blockIdx.x * 256 + threadIdx.x;          // [0, 256*48*32)
    if (idx >= (int)BOP_ELEMS) return;
    int lane = idx & 31;
    int t    = (idx >> 5) % T_STEPS;
    int tile = idx / (T_STEPS * 32);
    int m    = tile * 16 + (lane & 15);
    float m0 = memory[(size_t)m * (T_STEPS * 2) + t * 2 + 0];
    float m1 = memory[(size_t)m * (T_STEPS * 2) + t * 2 + 1];
    float2 val = (lane < 16) ? make_float2(m0, m1)
                             : make_float2(1.0f, m0 * m0 + m1 * m1);
    bop[idx] = val;
}

// ---------------------------------------------------------------------------
// Kernel 1: distance + argmin via V_WMMA_F32_16X16X4_F32
//   dist^2(b,m,t) = ||g||^2 + ||m||^2 - 2 g.m  as a 16x16x4 matmul:
//   A[b,:] = [-2g0, -2g1, ||g||^2, 1]   B[:,m] = [m0, m1, 1, ||m||^2]
//   NaN/negative-cancellation guard: max(sqrt_raw(x), 0) == sqrt(max(x,0))
//   since raw v_sqrt_f32(neg) = NaN and maximumNumber(NaN,0) = 0.
// ---------------------------------------------------------------------------
__global__ __launch_bounds__(256)
void cls_dist_argmin_kernel(const float* __restrict__ target,
                            const float* __restrict__ memory,
                            const float2* __restrict__ bop,
                            unsigned long long* __restrict__ packed,
                            int useBop) {
    __shared__ float2 ldsA2[T_STEPS * 32];       // [t][lane]: pre-selected A
    __shared__ unsigned long long ldsBest[16];

    const int tid   = threadIdx.x;
    const int wave  = tid >> 5;
    const int lane  = tid & 31;
    const int hi    = lane >> 4;                 // lane half: 0 or 1
    const int n     = lane & 15;                 // column / row index in tile
    const int bBase = blockIdx.y * 16;
    const int chunk = blockIdx.x;

    // Stage A operands pre-selected per lane slot: one ds_load_b64 per t later.
    for (int i = tid; i < T_STEPS * 32; i += 256) {
        int slot = i & 31;
        int t    = i >> 5;
        int b    = slot & 15;
        float g0 = target[(bBase + b) * (T_STEPS * 2) + t * 2 + 0];
        float g1 = target[(bBase + b) * (T_STEPS * 2) + t * 2 + 1];
        ldsA2[t * 32 + slot] = (slot < 16)
            ? make_float2(-2.0f * g0, -2.0f * g1)
            : make_float2(g0 * g0 + g1 * g1, 1.0f);
    }
    if (tid < 16) ldsBest[tid] = 0xFFFFFFFFFFFFFFFFULL;
    __syncthreads();

    float    best_d[8];
    unsigned best_m[8];
#pragma unroll
    for (int v = 0; v < 8; ++v) { best_d[v] = 3.4e38f; best_m[v] = 0u; }

    for (int ti = wave; ti < TILES_PER_CHUNK; ti += WAVES_PER_BLOCK) {
        const int mTile = chunk * TILES_PER_CHUNK + ti;
        const int m     = mTile * 16 + n;

        v8f acc = {0.f, 0.f, 0.f, 0.f, 0.f, 0.f, 0.f, 0.f};

        if (useBop) {
            // Fast path: per-lane B operand precomputed, coalesced loads.
            const float2* __restrict__ bp =
                bop + (size_t)mTile * (T_STEPS * 32) + lane;
#pragma unroll 4
            for (int t = 0; t < T_STEPS; ++t) {
                float2 av = ldsA2[t * 32 + lane];      // ds_load_b64
                float2 bw = bp[t * 32];                // global_load_b64
                v2f a;  a.x = av.x;  a.y = av.y;
                v2f bv; bv.x = bw.x; bv.y = bw.y;

                v8f c = {0.f, 0.f, 0.f, 0.f, 0.f, 0.f, 0.f, 0.f};
                c = __builtin_amdgcn_wmma_f32_16x16x4_f32(
                        false, a, false, bv, (short)0, c, false, false);
#pragma unroll
                for (int v = 0; v < 8; ++v) {
                    acc[v] += __builtin_fmaxf(
                        __builtin_amdgcn_sqrtf(c[v]), 0.0f);
                }
            }
        } else {
            // Fallback (small ws): build B operand inline from memory.
            const float* __restrict__ mp = memory + (size_t)m * (T_STEPS * 2);
#pragma unroll 4
            for (int t = 0; t < T_STEPS; ++t) {
                float2 av = ldsA2[t * 32 + lane];
                float m0 = mp[t * 2 + 0];
                float m1 = mp[t * 2 + 1];
                v2f a;  a.x = av.x;  a.y = av.y;
                v2f bv;
                bv.x = hi ? 1.0f : m0;
                bv.y = hi ? (m0 * m0 + m1 * m1) : m1;

                v8f c = {0.f, 0.f, 0.f, 0.f, 0.f, 0.f, 0.f, 0.f};
                c = __builtin_amdgcn_wmma_f32_16x16x4_f32(
                        false, a, false, bv, (short)0, c, false, false);
#pragma unroll
                for (int v = 0; v < 8; ++v) {
                    acc[v] += __builtin_fmaxf(
                        __builtin_amdgcn_sqrtf(c[v]), 0.0f);
                }
            }
        }

#pragma unroll
        for (int v = 0; v < 8; ++v) {
            if (acc[v] < best_d[v]) { best_d[v] = acc[v]; best_m[v] = (unsigned)m; }
        }
    }

    // D-layout: lane holds column N=n, VGPR v holds row M = v + 8*hi
#pragma unroll
    for (int v = 0; v < 8; ++v) {
        int bl = v + 8 * hi;
        unsigned long long pk =
            ((unsigned long long)__float_as_uint(best_d[v]) << 32) |
            (unsigned long long)best_m[v];
        atomicMin(&ldsBest[bl], pk);
    }
    __syncthreads();
    if (tid < 16) atomicMin(&packed[bBase + tid], ldsBest[tid]);
}

// ---------------------------------------------------------------------------
// Kernel 2: per-row logsumexp -> nll[b]; also emit true_index as float
// ---------------------------------------------------------------------------
__global__ __launch_bounds__(256)
void cls_nll_kernel(const float* __restrict__ preds,
                    const unsigned long long* __restrict__ packed,
                    float* __restrict__ nll,
                    float* __restrict__ out_idx) {
    __shared__ float red[256];
    const int b   = blockIdx.x;
    const int tid = threadIdx.x;
    const float* __restrict__ row = preds + (size_t)b * M_TOT;

    float vals[16];
    float mx = -3.4e38f;
#pragma unroll
    for (int i = 0; i < 16; ++i) {
        vals[i] = row[tid + i * 256];
        mx = fmaxf(mx, vals[i]);
    }
    red[tid] = mx; __syncthreads();
    for (int s = 128; s > 0; s >>= 1) {
        if (tid < s) red[tid] = fmaxf(red[tid], red[tid + s]);
        __syncthreads();
    }
    mx = red[0]; __syncthreads();

    float sum = 0.0f;
#pragma unroll
    for (int i = 0; i < 16; ++i) sum += expf(vals[i] - mx);
    red[tid] = sum; __syncthreads();
    for (int s = 128; s > 0; s >>= 1) {
        if (tid < s) red[tid] += red[tid + s];
        __syncthreads();
    }

    if (tid == 0) {
        unsigned idx = (unsigned)(packed[b] & 0xFFFFFFFFULL);
        float lse = mx + logf(red[0]);
        nll[b]     = lse - row[idx];
        out_idx[b] = (float)idx;
    }
}

// ---------------------------------------------------------------------------
// Kernel 3: mean of nll -> loss
// ---------------------------------------------------------------------------
__global__ void cls_loss_kernel(const float* __restrict__ nll,
                                float* __restrict__ out) {
    __shared__ float red[512];
    const int tid = threadIdx.x;
    red[tid] = nll[tid];
    __syncthreads();
    for (int s = 256; s > 0; s >>= 1) {
        if (tid < s) red[tid] += red[tid + s];
        __syncthreads();
    }
    if (tid == 0) out[0] = red[0] / (float)B_TOT;
}

// ---------------------------------------------------------------------------
extern "C" void kernel_launch(void* const* d_in, const int* in_sizes, int n_in,
                              void* d_out, int out_size, void* d_ws, size_t ws_size,
                              hipStream_t stream) {
    const float* preds  = (const float*)d_in[0];   // (512, 4096)
    const float* target = (const float*)d_in[1];   // (512, 1, 48, 2)
    const float* memory = (const float*)d_in[2];   // (4096, 48, 2)
    float* out = (float*)d_out;                    // [0]=loss, [1..512]=index

    unsigned long long* packed = (unsigned long long*)d_ws;       // 512 * 8B
    float* nll = (float*)((char*)d_ws + 4096);                    // 512 * 4B
    float2* bop = (float2*)((char*)d_ws + BOP_OFFSET);            // 3 MB

    size_t need = (size_t)BOP_OFFSET + BOP_ELEMS * sizeof(float2);
    int useBop = (ws_size >= need) ? 1 : 0;

    cls_init_kernel<<<1, 512, 0, stream>>>(packed);

    if (useBop) {
        int nb = (int)((BOP_ELEMS + 255) / 256);
        cls_bop_kernel<<<nb, 256, 0, stream>>>(memory, bop);
    }

    dim3 g1(CHUNKS, B_TOT / 16);
    cls_dist_argmin_kernel<<<g1, 256, 0, stream>>>(target, memory, bop,
                                                   packed, useBop);

    cls_nll_kernel<<<B_TOT, 256, 0, stream>>>(preds, packed, nll, out + 1);

    cls_loss_kernel<<<1, 512, 0, stream>>>(nll, out);
}